// NCC_HnN_30846455120060
// MI455X (gfx1250) — compile-verified
//
#include <hip/hip_runtime.h>

// ---------------------------------------------------------------------------
// Fused local-NCC (9x9x9 window) for MI455X / gfx1250.
//   - separable box sums: W-axis and H-axis passes done as banded GEMMs with
//     V_WMMA_F32_16X16X4_F32 (full fp32 precision), D-axis + NCC in VALU.
//   - one-wave (32-thread) workgroups, 16x16x16 output tile per wave.
//   - LDS: 5ch x 16 x 26 transpose buffer + 9-slab x 256-pos x 5ch ring.
//   - deterministic two-stage reduction (no float atomics).
// ---------------------------------------------------------------------------

typedef __attribute__((ext_vector_type(2))) float v2f;
typedef __attribute__((ext_vector_type(8))) float v8f;

#define DIN 160
#define HIN 192
#define WIN_ 160
#define DO_ 152
#define HO_ 184
#define WO_ 152
#define WINSZ 729.0f
#define NOUT 4251136.0f /* 152*184*152 */

__device__ __forceinline__ v8f wmma_k4(v2f a, v2f b, v8f c) {
  // D = A(16x4,f32) * B(4x16,f32) + C(16x16,f32)
  return __builtin_amdgcn_wmma_f32_16x16x4_f32(false, a, false, b, (short)0, c,
                                               false, false);
}

__global__ __launch_bounds__(32) void ncc_fused_kernel(
    const float* __restrict__ I, const float* __restrict__ J,
    float* __restrict__ partial) {
  __shared__ float P_lds[5 * 16 * 26];   // [ch][w'(16)][h(24,pad26)]
  __shared__ float ringb[9 * 256 * 5];   // [slot][pos(h'*16+w')][ch]

  const int lane = threadIdx.x;
  const int m = lane & 15;  // M index (filtered-axis output) / N index
  const int g = lane >> 4;  // half-wave select
  const int bw = blockIdx.x, bh = blockIdx.y, bd = blockIdx.z;
  const int w0 = min(bw * 16, WO_ - 16);
  const int h0 = min(bh * 16, HO_ - 16);
  const int d0 = min(bd * 16, DO_ - 16);

  // Constant banded A matrices: A_c[m][k] = 1 iff m <= 4c+k <= m+8.
  // A-layout (16x4 f32): lanes 0-15 hold M=lane {K0,K1}; lanes 16-31 {K2,K3}.
  v2f A[6];
#pragma unroll
  for (int c = 0; c < 6; ++c) {
    const int k0 = 4 * c + 2 * g;
    A[c].x = (k0 >= m && k0 <= m + 8) ? 1.0f : 0.0f;
    A[c].y = (k0 + 1 >= m && k0 + 1 <= m + 8) ? 1.0f : 0.0f;
  }

  float accsum = 0.0f;

  for (int s = d0; s < d0 + 24; ++s) {
    // ---- pass 1: W-axis 9-tap sums for 24 halo rows (two 16-row chains) ----
#pragma unroll
    for (int chain = 0; chain < 2; ++chain) {
      int hrow = h0 + 16 * chain + m;  // this lane's N = source row
      if (hrow > HIN - 1) hrow = HIN - 1;  // clamp (lanes discarded below)
      const int rowbase = (s * HIN + hrow) * WIN_;
      v8f a0 = {}, a1 = {}, a2 = {}, a3 = {}, a4 = {};
#pragma unroll
      for (int c = 0; c < 6; ++c) {
        const int wp = w0 + 4 * c + 2 * g;  // B-layout: K=2g,2g+1 pair
        const v2f iv = *(const v2f*)(I + rowbase + wp);
        const v2f jv = *(const v2f*)(J + rowbase + wp);
        v2f b2, b3, b4;
        b2.x = iv.x * iv.x; b2.y = iv.y * iv.y;
        b3.x = jv.x * jv.x; b3.y = jv.y * jv.y;
        b4.x = iv.x * jv.x; b4.y = iv.y * jv.y;
        a0 = wmma_k4(A[c], iv, a0);
        a1 = wmma_k4(A[c], jv, a1);
        a2 = wmma_k4(A[c], b2, a2);
        a3 = wmma_k4(A[c], b3, a3);
        a4 = wmma_k4(A[c], b4, a4);
      }
      // D-layout: element (v,g,lane): M = w' = v+8g, N = h = lane&15.
      // Store transposed into P_lds[ch][w'][h] (row stride 26: conflict-free).
      const int hl = 16 * chain + m;
      if (chain == 0 || m < 8) {
#pragma unroll
        for (int v = 0; v < 8; ++v) {
          const int wq = v + 8 * g;
          P_lds[0 * 416 + wq * 26 + hl] = a0[v];
          P_lds[1 * 416 + wq * 26 + hl] = a1[v];
          P_lds[2 * 416 + wq * 26 + hl] = a2[v];
          P_lds[3 * 416 + wq * 26 + hl] = a3[v];
          P_lds[4 * 416 + wq * 26 + hl] = a4[v];
        }
      }
    }
    __syncthreads();

    // ---- pass 2: H-axis 9-tap sums. M = h' out, N = w' = lane&15 ----
    {
      v8f c0 = {}, c1 = {}, c2 = {}, c3 = {}, c4 = {};
#pragma unroll
      for (int c = 0; c < 6; ++c) {
        const int hk = 4 * c + 2 * g;  // K pair for this lane
        const v2f b0 = *(const v2f*)&P_lds[0 * 416 + m * 26 + hk];
        const v2f b1 = *(const v2f*)&P_lds[1 * 416 + m * 26 + hk];
        const v2f b2 = *(const v2f*)&P_lds[2 * 416 + m * 26 + hk];
        const v2f b3 = *(const v2f*)&P_lds[3 * 416 + m * 26 + hk];
        const v2f b4 = *(const v2f*)&P_lds[4 * 416 + m * 26 + hk];
        c0 = wmma_k4(A[c], b0, c0);
        c1 = wmma_k4(A[c], b1, c1);
        c2 = wmma_k4(A[c], b2, c2);
        c3 = wmma_k4(A[c], b3, c3);
        c4 = wmma_k4(A[c], b4, c4);
      }
      // ring store: pos = h'*16 + w', channel-interleaved stride 5.
      const int slot = (s - d0) % 9;
#pragma unroll
      for (int v = 0; v < 8; ++v) {
        const int pos = (v + 8 * g) * 16 + m;
        const int base = slot * 1280 + pos * 5;
        ringb[base + 0] = c0[v];
        ringb[base + 1] = c1[v];
        ringb[base + 2] = c2[v];
        ringb[base + 3] = c3[v];
        ringb[base + 4] = c4[v];
      }
    }
    __syncthreads();

    // ---- pass 3: D-axis 9-tap sum over the ring + NCC + accumulate ----
    if (s >= d0 + 8) {
      const int dglob = s - 8;
      const bool dok = dglob >= bd * 16;  // ownership (overlapped edge tiles)
#pragma unroll
      for (int i = 0; i < 8; ++i) {
        const int pos = lane + 32 * i;
        const int hq = pos >> 4, wq = pos & 15;
        float S0 = 0.f, S1 = 0.f, S2 = 0.f, S3 = 0.f, S4 = 0.f;
#pragma unroll
        for (int t = 0; t < 9; ++t) {
          const int base = t * 1280 + pos * 5;
          S0 += ringb[base + 0];
          S1 += ringb[base + 1];
          S2 += ringb[base + 2];
          S3 += ringb[base + 3];
          S4 += ringb[base + 4];
        }
        if (dok && (h0 + hq >= bh * 16) && (w0 + wq >= bw * 16)) {
          const float uI = S0 / WINSZ;
          const float uJ = S1 / WINSZ;
          const float cross = S4 - uJ * S0 - uI * S1 + uI * uJ * WINSZ;
          const float Ivar = S2 - 2.0f * uI * S0 + uI * uI * WINSZ;
          const float Jvar = S3 - 2.0f * uJ * S1 + uJ * uJ * WINSZ;
          accsum += cross * cross / (Ivar * Jvar + 1e-5f);
        }
      }
    }
    __syncthreads();
  }

  // wave32 reduction, lane 0 writes this block's partial sum.
#pragma unroll
  for (int off = 16; off > 0; off >>= 1) accsum += __shfl_xor(accsum, off, 32);
  if (lane == 0)
    partial[(bd * gridDim.y + bh) * gridDim.x + bw] = accsum;
}

__global__ __launch_bounds__(256) void ncc_reduce_kernel(
    const float* __restrict__ partial, int n, float* __restrict__ out) {
  __shared__ float sm[256];
  float s = 0.0f;
  for (int i = threadIdx.x; i < n; i += 256) s += partial[i];
  sm[threadIdx.x] = s;
  __syncthreads();
  for (int off = 128; off > 0; off >>= 1) {
    if (threadIdx.x < (unsigned)off) sm[threadIdx.x] += sm[threadIdx.x + off];
    __syncthreads();
  }
  if (threadIdx.x == 0) out[0] = -sm[0] / NOUT;
}

extern "C" void kernel_launch(void* const* d_in, const int* in_sizes, int n_in,
                              void* d_out, int out_size, void* d_ws,
                              size_t ws_size, hipStream_t stream) {
  (void)in_sizes; (void)n_in; (void)out_size; (void)ws_size;
  const float* I = (const float*)d_in[0];  // "out" in reference
  const float* J = (const float*)d_in[1];  // "y_pred"
  float* partial = (float*)d_ws;           // 1200 floats
  dim3 grid(10, 12, 10);                   // ceil(152/16), ceil(184/16), ceil(152/16)
  ncc_fused_kernel<<<grid, 32, 0, stream>>>(I, J, partial);
  ncc_reduce_kernel<<<1, 256, 0, stream>>>(partial, 10 * 12 * 10,
                                           (float*)d_out);
}